// SphericalBasisLayer_74165495267558
// MI455X (gfx1250) — compile-verified
//
#include <hip/hip_runtime.h>
#include <hip/hip_bf16.h>
#include <stdint.h>

#define NUM_SPH 7
#define NUM_RAD 6
#define NCOL    42
#define BLOCK   256
#define TILE_FLOATS (BLOCK * NCOL)      // 10752 floats per block tile
#define TILE_VEC4   (TILE_FLOATS / 4)   // 2688 float4 per block tile

// ---------------- compile-time tables (constexpr, folded into literals) ----
namespace tbl {
constexpr double PI = 3.14159265358979323846264338327950288;

// First 6 positive zeros of spherical Bessel j_l, l = 0..6
constexpr double Z[NUM_SPH][NUM_RAD] = {
  { 3.141592653589793,  6.283185307179586,  9.424777960769379, 12.566370614359172, 15.707963267948966, 18.849555921538759},
  { 4.493409457909064,  7.725251836937707, 10.904121659428899, 14.066193912831473, 17.220755271930768, 20.371302959287563},
  { 5.763459196894550,  9.095011330476355, 12.322940970566582, 15.514603010886749, 18.689036355362822, 21.853874222709884},
  { 6.987932000500519, 10.417118547379365, 13.698023153250246, 16.923621285214318, 20.121806174454748, 23.304246988939617},
  { 8.182561452571243, 11.704907154570419, 15.039664707615553, 18.301255959541963, 21.525417733399804, 24.727565547835880},
  { 9.355812111042747, 12.966530172775206, 16.354709639341357, 19.653152101821259, 22.904550647902631, 26.127750137366465},
  {10.512835408093997, 14.207392458842874, 17.647974913172741, 20.983463068944005, 24.262768059091771, 27.507868340976654}
};

constexpr double csin(double x) {
  double y = x;
  while (y >  PI) y -= 2.0 * PI;
  while (y < -PI) y += 2.0 * PI;
  double y2 = y * y, term = y, sum = y;
  for (int n = 1; n <= 13; ++n) { term *= -y2 / double((2*n) * (2*n + 1)); sum += term; }
  return sum;
}
constexpr double ccos(double x) {
  double y = x;
  while (y >  PI) y -= 2.0 * PI;
  while (y < -PI) y += 2.0 * PI;
  double y2 = y * y, term = 1.0, sum = 1.0;
  for (int n = 1; n <= 13; ++n) { term *= -y2 / double((2*n - 1) * (2*n)); sum += term; }
  return sum;
}
constexpr double csqrt(double x) {
  double g = x > 1.0 ? x : 1.0;
  for (int i = 0; i < 64; ++i) g = 0.5 * (g + x / g);
  return g;
}
constexpr double cjn(double x, int l) {
  double j0 = csin(x) / x;
  if (l == 0) return j0;
  double j1 = csin(x) / (x * x) - ccos(x) / x;
  for (int m = 1; m < l; ++m) { double t = double(2*m + 1) / x * j1 - j0; j0 = j1; j1 = t; }
  return j1;
}

struct Tables {
  float z[NUM_SPH][NUM_RAD];
  float nrm[NUM_SPH][NUM_RAD];
  float sph[NUM_SPH];
};
constexpr Tables make_tables() {
  Tables t{};
  for (int l = 0; l < NUM_SPH; ++l) {
    for (int i = 0; i < NUM_RAD; ++i) {
      t.z[l][i] = (float)Z[l][i];
      double jv = cjn(Z[l][i], l + 1);
      t.nrm[l][i] = (float)(1.0 / csqrt(0.5 * jv * jv));
    }
    t.sph[l] = (float)csqrt((2.0 * l + 1.0) / (4.0 * PI));
  }
  return t;
}
constexpr Tables TAB = make_tables();
} // namespace tbl

// ---------------- gfx1250 async LDS -> global store -----------------------
// Probe round 1 showed the builtin exists and expects int4 pointers:
//   param 0: 'int __vector(4) __device__ *'  (address_space(1) == __device__)
typedef int v4i __attribute__((ext_vector_type(4)));
typedef __attribute__((address_space(1))) v4i* g4p_t;   // global int4*
typedef __attribute__((address_space(3))) v4i* l4p_t;   // LDS int4*
typedef __attribute__((address_space(3))) void LAS;

__device__ __forceinline__ void async_store_b128(float* gptr, float* lptr) {
#if __has_builtin(__builtin_amdgcn_global_store_async_from_lds_b128)
  __builtin_amdgcn_global_store_async_from_lds_b128((g4p_t)gptr, (l4p_t)lptr, 0, 0);
#else
  asm volatile("global_store_async_from_lds_b128 %0, %1, off"
               :
               : "v"((uint64_t)(uintptr_t)gptr),
                 "v"((uint32_t)(uintptr_t)(LAS*)lptr)
               : "memory");
#endif
}

__device__ __forceinline__ void wait_async_zero() {
#if __has_builtin(__builtin_amdgcn_s_wait_asynccnt)
  __builtin_amdgcn_s_wait_asynccnt(0);
#else
  asm volatile("s_wait_asynccnt 0" ::: "memory");
#endif
}

// ---------------- fused kernel: one thread per triplet ---------------------
__global__ __launch_bounds__(BLOCK)
void sph_basis_kernel(const float* __restrict__ d,
                      const float* __restrict__ angles,
                      const int*   __restrict__ bond,
                      float* __restrict__ out,
                      int T)
{
  __shared__ float sdata[TILE_FLOATS];

  const int tid = threadIdx.x;
  const int t   = blockIdx.x * BLOCK + tid;
  const int tc  = (t < T) ? t : (T - 1);   // clamp tail, never early-return

  // ---- angular part: Legendre P_l(cos(angle)) * sph_norm[l]
  const float cv = __cosf(angles[tc]);
  float cb[NUM_SPH];
  cb[0] = tbl::TAB.sph[0];
  cb[1] = tbl::TAB.sph[1] * cv;
  {
    float pm2 = 1.0f, pm1 = cv;
#pragma unroll
    for (int l = 2; l < NUM_SPH; ++l) {
      float p = (float(2*l - 1) * cv * pm1 - float(l - 1) * pm2) * (1.0f / float(l));
      cb[l] = tbl::TAB.sph[l] * p;
      pm2 = pm1; pm1 = p;
    }
  }

  // ---- radial part: envelope(d_scaled)
  const int   e   = bond[tc];
  const float ds  = d[e] * 0.2f;                    // d / CUTOFF
  const float rds = __builtin_amdgcn_rcpf(ds);
  const float ds2 = ds * ds;
  const float ds4 = ds2 * ds2;
  const float ds5 = ds4 * ds;
  const float ds6 = ds5 * ds;
  const float ds7 = ds6 * ds;
  float env = rds - 28.0f * ds5 + 48.0f * ds6 - 21.0f * ds7;  // p=6: a=-28,b=48,c=-21
  env = (ds < 1.0f) ? env : 0.0f;

  // ---- 42 spherical-Bessel basis values -> LDS row
  float* row = &sdata[tid * NCOL];
#pragma unroll
  for (int l = 0; l < NUM_SPH; ++l) {
    const float fl = env * cb[l];
#pragma unroll
    for (int i = 0; i < NUM_RAD; ++i) {
      const float x  = ds * tbl::TAB.z[l][i];
      const float rx = __builtin_amdgcn_rcpf(x);
      const float s  = __sinf(x);
      float j;
      if (l == 0) {
        j = s * rx;
      } else {
        const float c = __cosf(x);
        float j0 = s * rx;
        float j1 = (s * rx - c) * rx;
#pragma unroll
        for (int m = 1; m < l; ++m) {
          float jn = float(2*m + 1) * rx * j1 - j0;
          j0 = j1; j1 = jn;
        }
        j = j1;
      }
      row[l * NUM_RAD + i] = fl * tbl::TAB.nrm[l][i] * j;
    }
  }

  __syncthreads();   // all LDS rows visible before the DMA reads them

  // ---- stream the tile LDS -> global via async b128 stores (coalesced)
  const size_t totalF = (size_t)T * NCOL;
  const size_t total4 = totalF / 4;                 // guaranteed whole float4s valid
  const size_t base4  = (size_t)blockIdx.x * TILE_VEC4;
#pragma unroll
  for (int k = 0; k < (TILE_VEC4 + BLOCK - 1) / BLOCK; ++k) {   // 11 iters, last partial
    const int c4 = k * BLOCK + tid;
    if (c4 < TILE_VEC4) {
      const size_t g4 = base4 + c4;
      if (g4 < total4) {
        async_store_b128(out + g4 * 4, &sdata[c4 * 4]);
      }
    }
  }

  // scalar cleanup for a non-multiple-of-4 total (not hit for T=2M, kept safe)
  const size_t flushed = total4 * 4;
  if (blockIdx.x == (unsigned)gridDim.x - 1 && flushed < totalF) {
    for (size_t f = flushed + tid; f < totalF; f += BLOCK) {
      const size_t local = f - (size_t)blockIdx.x * TILE_FLOATS;
      out[f] = sdata[local];
    }
  }

  wait_async_zero();  // LDS must stay live until the async DMA has read it
}

// ---------------- launch ---------------------------------------------------
extern "C" void kernel_launch(void* const* d_in, const int* in_sizes, int n_in,
                              void* d_out, int out_size, void* d_ws, size_t ws_size,
                              hipStream_t stream) {
  (void)n_in; (void)out_size; (void)d_ws; (void)ws_size;
  const float* d      = (const float*)d_in[0];
  const float* angles = (const float*)d_in[1];
  const int*   bond   = (const int*)d_in[2];
  float*       out    = (float*)d_out;
  const int T = in_sizes[1];
  const int blocks = (T + BLOCK - 1) / BLOCK;
  hipLaunchKernelGGL(sph_basis_kernel, dim3(blocks), dim3(BLOCK), 0, stream,
                     d, angles, bond, out, T);
}